// FewShotSegmentation_35579509080788
// MI455X (gfx1250) — compile-verified
//
#include <hip/hip_runtime.h>
#include <hip/hip_bf16.h>
#include <math.h>

typedef float v2f __attribute__((ext_vector_type(2)));
typedef float v8f __attribute__((ext_vector_type(8)));

#define HW 16384      // 128*128
#define C  512
#define B  2
#define EPSV 1e-8f

// ---------------------------------------------------------------------------
// Kernel 1: bilinear 4x downsample of masks (512x512 -> 128x128, half-pixel,
// no antialias => average of the 2x2 interior samples) + per-batch class counts.
// ---------------------------------------------------------------------------
__global__ void k_mask_resize(const float* __restrict__ s_masks,
                              float* __restrict__ mask_r,
                              float* __restrict__ counts) {
    int idx = blockIdx.x * blockDim.x + threadIdx.x;   // 0 .. 2*16384
    int b   = idx >> 14;
    int rem = idx & (HW - 1);
    int y = rem >> 7, x = rem & 127;
    const float* mb = s_masks + (size_t)b * 512 * 512;
    int yy = 4 * y, xx = 4 * x;
    float v = 0.25f * (mb[(yy + 1) * 512 + xx + 1] + mb[(yy + 1) * 512 + xx + 2] +
                       mb[(yy + 2) * 512 + xx + 1] + mb[(yy + 2) * 512 + xx + 2]);
    mask_r[idx] = v;
    atomicAdd(&counts[b],     (v == 1.0f) ? 1.0f : 0.0f);   // class-1 count
    atomicAdd(&counts[2 + b], (v == 0.0f) ? 1.0f : 0.0f);   // class-0 count
}

// ---------------------------------------------------------------------------
// Kernel 2: prototypes. One block per (b, c); masked sums over 16384 pixels.
// proto layout: (class, batch, channel) flattened.
// ---------------------------------------------------------------------------
__global__ void k_proto(const float* __restrict__ s_features,
                        const float* __restrict__ mask_r,
                        const float* __restrict__ counts,
                        float* __restrict__ proto_ws,
                        float* __restrict__ proto_out) {
    int bid = blockIdx.x;           // 0 .. 1023
    int b = bid >> 9;
    int c = bid & (C - 1);
    const float* f = s_features + ((size_t)(b * C + c)) * HW;
    const float* m = mask_r + (size_t)b * HW;

    float s1 = 0.f, s0 = 0.f;
    for (int i = threadIdx.x; i < HW; i += 256) {
        float fv = f[i], mv = m[i];
        if (mv == 1.0f)      s1 += fv;
        else if (mv == 0.0f) s0 += fv;
    }
    __shared__ float r1[256], r0[256];
    r1[threadIdx.x] = s1; r0[threadIdx.x] = s0;
    __syncthreads();
    for (int st = 128; st > 0; st >>= 1) {
        if (threadIdx.x < st) {
            r1[threadIdx.x] += r1[threadIdx.x + st];
            r0[threadIdx.x] += r0[threadIdx.x + st];
        }
        __syncthreads();
    }
    if (threadIdx.x == 0) {
        float p1 = r1[0] / counts[b];
        float p0 = r0[0] / counts[2 + b];
        int i0 = 0 * (B * C) + b * C + c;   // class 0
        int i1 = 1 * (B * C) + b * C + c;   // class 1
        proto_ws[i0] = p0;  proto_ws[i1] = p1;
        proto_out[i0] = p0; proto_out[i1] = p1;
    }
}

// ---------------------------------------------------------------------------
// Kernel 3: prototype norms. 4 waves, one (class,batch) pair each.
// ---------------------------------------------------------------------------
__global__ void k_pn(const float* __restrict__ proto_ws, float* __restrict__ pn) {
    int w = threadIdx.x >> 5;    // pair index = class*2 + b
    int lane = threadIdx.x & 31;
    float s = 0.f;
    for (int k = lane; k < C; k += 32) {
        float p = proto_ws[w * C + k];
        s += p * p;
    }
    for (int off = 16; off > 0; off >>= 1) s += __shfl_xor(s, off, 32);
    if (lane == 0) pn[w] = sqrtf(s);
}

// ---------------------------------------------------------------------------
// Kernel 4: matching via v_wmma_f32_16x16x4_f32.
// One wave per 16-spatial tile; K iterates over 512 channels in steps of 4.
// A (16x4): lanes 0-15 hold K={k0,k0+1}, lanes 16-31 hold K={k0+2,k0+3}.
// B (4x16): column N=0 -> proto batch 0, N=1 -> proto batch 1, rest zero.
// Sum of squares of q accumulated from the same A registers (q read once).
// ---------------------------------------------------------------------------
__global__ void k_match(const float* __restrict__ q,
                        const float* __restrict__ proto_ws,
                        const float* __restrict__ pn,
                        float* __restrict__ out_logits,
                        float* __restrict__ out_sim) {
    __shared__ float lp[2 * B * C];     // prototypes (class,batch,channel)
    __shared__ float lpn[4];
    __shared__ float lt[4][272];        // per-wave 16x16 D tile + 16 sumsq

    int tid = threadIdx.x;
    for (int i = tid; i < 2 * B * C; i += 128) lp[i] = proto_ws[i];
    if (tid < 4) lpn[tid] = pn[tid];
    __syncthreads();

    int wv   = tid >> 5;
    int lane = tid & 31;
    int half = lane >> 4;        // 0: K pair {0,1}; 1: K pair {2,3}
    int m    = lane & 15;        // A: row M / B: col N / D: col N

    int w  = blockIdx.x * 4 + wv;       // global wave/tile id: 0..2047
    int a  = w >> 10;                   // q-batch (== prototype class index)
    int t  = w & 1023;
    int s0 = t << 4;

    const float* qa = q + (size_t)a * C * HW + s0 + m;
    const float* lpa = lp + a * (B * C);    // prototypes for class a

    v8f acc = {};
    float ssq = 0.f;
    int n = m;
    int np = (n < 2) ? n : 0;           // clamp LDS index, select below

    for (int k0 = 0; k0 < C; k0 += 4) {
        int cA = k0 + half * 2;
        float a0 = qa[(size_t)cA * HW];
        float a1 = qa[(size_t)(cA + 1) * HW];
        ssq += a0 * a0 + a1 * a1;

        float lv0 = lpa[np * C + cA];
        float lv1 = lpa[np * C + cA + 1];
        float b0 = (n < 2) ? lv0 : 0.0f;
        float b1 = (n < 2) ? lv1 : 0.0f;

        if (k0 + 8 < C)
            __builtin_prefetch((const void*)(qa + (size_t)(k0 + 8) * HW), 0, 1);

        v2f A = {a0, a1};
        v2f Bv = {b0, b1};
        acc = __builtin_amdgcn_wmma_f32_16x16x4_f32(
            /*neg_a=*/false, A, /*neg_b=*/false, Bv,
            /*c_mod=*/(short)0, acc, /*reuse_a=*/false, /*reuse_b=*/false);
    }

    // Fold sumsq across the two K-halves holding the same spatial row M.
    float ssqf = ssq + __shfl_xor(ssq, 16, 32);

    // D layout: VGPR v, lanes 0-15 -> M=v, lanes 16-31 -> M=v+8, N=lane&15.
#pragma unroll
    for (int v = 0; v < 8; ++v)
        lt[wv][(v + half * 8) * 16 + m] = acc[v];
    if (half == 0) lt[wv][256 + m] = ssqf;
    __syncthreads();

    if (lane < 16) {
        int s = s0 + lane;
        float num0 = lt[wv][lane * 16 + 0];
        float num1 = lt[wv][lane * 16 + 1];
        float qn = sqrtf(lt[wv][256 + lane]);
        float d0 = fmaxf(qn * lpn[a * 2 + 0], EPSV);
        float d1 = fmaxf(qn * lpn[a * 2 + 1], EPSV);
        float sim0 = num0 / d0;
        float sim1 = num1 / d1;
        out_sim[(size_t)a * 2 * HW + s]      = sim0;   // sim[a, 0, s]
        out_sim[(size_t)a * 2 * HW + HW + s] = sim1;   // sim[a, 1, s]
        // jnp.argmax takes first max on ties -> 1 only if strictly greater
        out_logits[(size_t)a * HW + s] = (sim1 > sim0) ? 1.0f : 0.0f;
    }
}

// ---------------------------------------------------------------------------
extern "C" void kernel_launch(void* const* d_in, const int* in_sizes, int n_in,
                              void* d_out, int out_size, void* d_ws, size_t ws_size,
                              hipStream_t stream) {
    const float* s_features = (const float*)d_in[0];   // (2,512,128,128)
    const float* s_masks    = (const float*)d_in[1];   // (2,1,512,512)
    const float* q_features = (const float*)d_in[2];   // (2,512,128,128)

    float* wsf    = (float*)d_ws;
    float* maskR  = wsf;               // 2*16384
    float* counts = wsf + 2 * HW;      // 4: count1[b0..b1], count0[b0..b1]
    float* proto  = counts + 4;        // 2*2*512
    float* pn     = proto + 2 * B * C; // 4

    float* o          = (float*)d_out;
    float* out_logits = o;                       // (2,128,128) as floats
    float* out_sim    = o + B * HW;              // (2,2,16384)
    float* out_proto  = out_sim + 2 * B * HW;    // (2,2,512)

    hipMemsetAsync(counts, 0, 4 * sizeof(float), stream);

    k_mask_resize<<<(2 * HW) / 256, 256, 0, stream>>>(s_masks, maskR, counts);
    k_proto<<<B * C, 256, 0, stream>>>(s_features, maskR, counts, proto, out_proto);
    k_pn<<<1, 128, 0, stream>>>(proto, pn);
    k_match<<<(2 * HW / 16) / 4, 128, 0, stream>>>(q_features, proto, pn,
                                                   out_logits, out_sim);
}